// LearnableDemosaick_17712445129210
// MI455X (gfx1250) — compile-verified
//
#include <hip/hip_runtime.h>

// Space-variant learnable demosaicking (green channel) for MI455X / gfx1250.
//
// GEMM formulation per 16-pixel tile:  D[16 pix x 16 ch] = A[16 x 28] * B[28 x 16]
//   A = im2col of the 5x5 edge-padded window (25 taps, zero-padded to K=28)
//   B = [sel_filts | green_filts] flattened taps (zero-padded rows 25..27)
// executed as 7 chained V_WMMA_F32_16X16X4_F32 accumulations (full fp32,
// matching the fp32 reference).
//
// Epilogue: the D tile is transposed through LDS so each of 128 threads owns
// one pixel's 16 channel values (4x ds_load_b128) and runs the softmax-dot
// serially — no redundant cross-lane work, fully coalesced stores.

typedef __attribute__((ext_vector_type(2))) float v2f;
typedef __attribute__((ext_vector_type(8))) float v8f;

#define NTAPS 25
#define NFILT 8
#define TILE_W 128          // 8 waves * 16 pixels per block row
#define ROWS_PER_BLOCK 8
#define IN_STRIDE 136       // TILE_W + 4 halo, padded
#define D_STRIDE 20         // 16 channels padded to 20 floats (80B, 16B-aligned)

__global__ __launch_bounds__(256)
void demosaick_green_wmma(const float* __restrict__ mosaick,
                          const float* __restrict__ sel_filts,
                          const float* __restrict__ green_filts,
                          float* __restrict__ out,
                          int H, int W) {
  __shared__ float s_in[5 * IN_STRIDE];
  __shared__ float s_d[8 * 16 * D_STRIDE];   // per wave: [pixel m][channel n]

  const int tid  = threadIdx.x;
  const int lane = tid & 31;
  const int wid  = tid >> 5;
  const int n8   = lane & 15;   // A row (pixel-in-tile) == B/D column (channel)
  const int hi   = lane >> 4;   // lane half: selects K pair (A/B) / M+8 (D)
  const int woff = wid * 16;    // wave's pixel offset within the strip

  const int x0 = blockIdx.x * TILE_W;
  const int y0 = blockIdx.y * ROWS_PER_BLOCK;
  const int b  = blockIdx.z;
  const float* src = mosaick + (size_t)b * H * W;
  float*       dst = out     + (size_t)b * H * W;

  // ---- B operand: [28 x 16] filter matrix, resident in registers ------
  // 32-bit B 4x16 layout mirrors A 16x4: lanes 0-15 hold rows K={4kb,4kb+1}
  // in the vgpr pair, lanes 16-31 hold K={4kb+2,4kb+3}.
  v2f Bv[7];
#pragma unroll
  for (int kb = 0; kb < 7; ++kb) {
    int k0 = kb * 4 + hi * 2;
    float b0 = 0.f, b1 = 0.f;
    if (k0 < NTAPS)
      b0 = (n8 < NFILT) ? sel_filts[n8 * NTAPS + k0]
                        : green_filts[(n8 - NFILT) * NTAPS + k0];
    if (k0 + 1 < NTAPS)
      b1 = (n8 < NFILT) ? sel_filts[n8 * NTAPS + k0 + 1]
                        : green_filts[(n8 - NFILT) * NTAPS + k0 + 1];
    Bv[kb].x = b0;
    Bv[kb].y = b1;
  }

  for (int ry = 0; ry < ROWS_PER_BLOCK; ++ry) {
    const int y = y0 + ry;

    // ---- Stage 5 edge-clamped input rows, x0-2 .. x0+TILE_W+1 --------
    for (int idx = tid; idx < 5 * (TILE_W + 4); idx += 256) {
      int r  = idx / (TILE_W + 4);
      int c  = idx - r * (TILE_W + 4);
      int gy = min(max(y - 2 + r, 0), H - 1);
      int gx = min(max(x0 - 2 + c, 0), W - 1);
      s_in[r * IN_STRIDE + c] = src[gy * W + gx];
    }
    __syncthreads();

    // ---- 16x28x16 GEMM via 7 chained f32 WMMAs -----------------------
    v8f acc = {};
#pragma unroll
    for (int kb = 0; kb < 7; ++kb) {
      int k0  = kb * 4 + hi * 2;
      int kc0 = min(k0, NTAPS - 1);
      int kc1 = min(k0 + 1, NTAPS - 1);
      // tap k -> (dy = k/5, dx = k%5); A(m,k) = s_in[dy][woff + m + dx]
      float a0 = s_in[(kc0 / 5) * IN_STRIDE + woff + n8 + (kc0 % 5)];
      float a1 = s_in[(kc1 / 5) * IN_STRIDE + woff + n8 + (kc1 % 5)];
      v2f a;
      a.x = (k0     < NTAPS) ? a0 : 0.f;
      a.y = (k0 + 1 < NTAPS) ? a1 : 0.f;
      acc = __builtin_amdgcn_wmma_f32_16x16x4_f32(
          false, a, false, Bv[kb], (short)0, acc, false, false);
    }

    // ---- Transpose D to LDS: s_d[wave][pixel m][channel n] -----------
    // D layout: VGPR v, lanes 0-15 -> pixel M=v, lanes 16-31 -> M=v+8;
    // lane%16 -> channel. Lanes 0-15 / 16-31 write disjoint rows.
#pragma unroll
    for (int v = 0; v < 8; ++v)
      s_d[wid * (16 * D_STRIDE) + (hi * 8 + v) * D_STRIDE + n8] = acc[v];
    __syncthreads();

    // ---- One pixel per thread: serial softmax(sel) . interp ----------
    if (tid < TILE_W) {
      const int p = tid;                 // pixel within the 128-px strip
      const int w = p >> 4, m = p & 15;
      const float4* ch =
          (const float4*)&s_d[w * (16 * D_STRIDE) + m * D_STRIDE];
      float4 c0 = ch[0], c1 = ch[1];     // sel scores   (channels 0-7)
      float4 c2 = ch[2], c3 = ch[3];     // green candidates (8-15)

      float mx = fmaxf(fmaxf(fmaxf(c0.x, c0.y), fmaxf(c0.z, c0.w)),
                       fmaxf(fmaxf(c1.x, c1.y), fmaxf(c1.z, c1.w)));
      float e0 = __expf(c0.x - mx), e1 = __expf(c0.y - mx);
      float e2 = __expf(c0.z - mx), e3 = __expf(c0.w - mx);
      float e4 = __expf(c1.x - mx), e5 = __expf(c1.y - mx);
      float e6 = __expf(c1.z - mx), e7 = __expf(c1.w - mx);
      float den = ((e0 + e1) + (e2 + e3)) + ((e4 + e5) + (e6 + e7));
      float num = e0 * c2.x;
      num = fmaf(e1, c2.y, num);
      num = fmaf(e2, c2.z, num);
      num = fmaf(e3, c2.w, num);
      num = fmaf(e4, c3.x, num);
      num = fmaf(e5, c3.y, num);
      num = fmaf(e6, c3.z, num);
      num = fmaf(e7, c3.w, num);
      float gh = num / den;

      int x = x0 + p;
      float mos = s_in[2 * IN_STRIDE + p + 2];        // center tap = mosaick
      float res = (((x ^ y) & 1) == 0) ? mos : gh;    // GRBG green sites
      dst[(size_t)y * W + x] = res;
    }
    __syncthreads();
  }
}

extern "C" void kernel_launch(void* const* d_in, const int* in_sizes, int n_in,
                              void* d_out, int out_size, void* d_ws, size_t ws_size,
                              hipStream_t stream) {
  (void)n_in; (void)out_size; (void)d_ws; (void)ws_size;
  const float* mosaick     = (const float*)d_in[0];
  const float* sel_filts   = (const float*)d_in[1];
  const float* green_filts = (const float*)d_in[2];
  float* out = (float*)d_out;

  const int H = 1024, W = 1024;
  const int Bn = in_sizes[0] / (H * W);   // = 8

  dim3 grid(W / TILE_W, H / ROWS_PER_BLOCK, Bn);
  demosaick_green_wmma<<<grid, dim3(256), 0, stream>>>(
      mosaick, sel_filts, green_filts, out, H, W);
}